// AutoCorrelationBlock_4801773437281
// MI455X (gfx1250) — compile-verified
//
#include <hip/hip_runtime.h>

// ---------------------------------------------------------------------------
// AutoCorrelation block for MI455X (gfx1250, wave32).
//   q = xW + b                         (WMMA f32 16x16x4, 16x64 reg-blocked)
//   corr[b,t] = (1/D) * circular-diagonal sums of Q Q^T
//                                      (WMMA f32 16x16x4, 64x64 reg-blocked)
//   delays = top-12(corr);  out = x + mean_k q[(t+delay_k) % T]
// ---------------------------------------------------------------------------

typedef float v2f __attribute__((ext_vector_type(2)));
typedef float v8f __attribute__((ext_vector_type(8)));

#define B_    8
#define T_    4096
#define D_    512
#define KTOP  12
#define TMASK (T_ - 1)
#define NEG_INF (-3.4028234663852886e38f)

static __device__ __forceinline__ v8f wmma_f32_16x16x4(v2f a, v2f b, v8f c) {
    // 8 args: (neg_a, A, neg_b, B, c_mod, C, reuse_a, reuse_b)
    return __builtin_amdgcn_wmma_f32_16x16x4_f32(
        false, a, false, b, (short)0, c, false, false);
}

// ---------------------------------------------------------------------------
// K0: zero the corr accumulator (workspace is poisoned by harness).
// ---------------------------------------------------------------------------
__global__ void k_zero_corr(float* __restrict__ corr) {
    int i = blockIdx.x * blockDim.x + threadIdx.x;
    if (i < B_ * T_) corr[i] = 0.0f;
}

// ---------------------------------------------------------------------------
// K1: q[32768, 512] = x[32768, 512] @ W[512, 512] + bias[512]
// 16x64 tile per wave: A frag loaded once per k-step, reused by 4 WMMAs.
// ---------------------------------------------------------------------------
__global__ void k_proj_gemm(const float* __restrict__ x,
                            const float* __restrict__ W,
                            const float* __restrict__ bias,
                            float* __restrict__ q) {
    const int lane  = threadIdx.x & 31;
    const int wave  = threadIdx.x >> 5;
    const int wtile = blockIdx.x * 8 + wave;   // 16384 wave-tiles total
    const int r0    = (wtile >> 3) * 16;       // 2048 row tiles  (M = 32768)
    const int c0    = (wtile & 7) * 64;        // 8 col supertiles (N = 512)
    const int m     = lane & 15;               // A row / B col / C col
    const int half  = lane >> 4;

    const float* xrow = x + (size_t)(r0 + m) * D_;
    v8f acc[4] = {};
#pragma unroll 1
    for (int k0 = 0; k0 < D_; k0 += 4) {
        v2f a;
        a.x = xrow[k0 + 2 * half];             // K = k0 + 2*half (+1)
        a.y = xrow[k0 + 2 * half + 1];
        const float* wk = W + (size_t)(k0 + 2 * half) * D_ + c0 + m;
#pragma unroll
        for (int jb = 0; jb < 4; ++jb) {
            v2f b;
            b.x = wk[16 * jb];
            b.y = wk[16 * jb + D_];
            acc[jb] = wmma_f32_16x16x4(a, b, acc[jb]);
        }
    }

#pragma unroll
    for (int jb = 0; jb < 4; ++jb) {
        const float bv = bias[c0 + 16 * jb + m];
        float* qout = q + (size_t)r0 * D_ + (c0 + 16 * jb + m);
#pragma unroll
        for (int v = 0; v < 8; ++v) {
            const int mm = v + 8 * half;       // C/D: vgpr v -> M = v (+8 hi half)
            qout[(size_t)mm * D_] = acc[jb][v] + bv;
        }
    }
}

// ---------------------------------------------------------------------------
// K2: circular-autocorrelation via Gram tiles, 64x64 register-blocked.
// Block = (i-supertile of 64 rows, batch b). 8 waves sweep the 64 j-super-
// tiles; each wave holds a 64x64 tile of G = Q Q^T in 16 v8f accumulators.
// Per k-step: 4 A frags + 4 B frags feed 16 WMMAs (1 dword/lane per WMMA).
// C[m,n] scattered into block-local LDS corr at t = (col - row) mod T.
// ---------------------------------------------------------------------------
__global__ void k_gram_corr(const float* __restrict__ q,
                            float* __restrict__ corr) {
    __shared__ float cl[T_];                   // 16 KB block-local corr slice
    const int b    = blockIdx.y;
    const int i0   = blockIdx.x * 64;
    const int lane = threadIdx.x & 31;
    const int wave = threadIdx.x >> 5;
    const int m    = lane & 15;
    const int half = lane >> 4;

    for (int t = threadIdx.x; t < T_; t += blockDim.x) cl[t] = 0.0f;
    __syncthreads();

    const float* Q = q + (size_t)b * T_ * D_;
    const float* arow[4];
#pragma unroll
    for (int ia = 0; ia < 4; ++ia)
        arow[ia] = Q + (size_t)(i0 + 16 * ia + m) * D_;

    for (int js = wave; js < T_ / 64; js += 8) {
        const int j0 = js * 64;
        const float* brow[4];
#pragma unroll
        for (int jb = 0; jb < 4; ++jb)
            brow[jb] = Q + (size_t)(j0 + 16 * jb + m) * D_;  // B[k,n]=Q[col,k]

        v8f acc[4][4] = {};
#pragma unroll 1
        for (int k0 = 0; k0 < D_; k0 += 4) {
            v2f a[4], bb[4];
#pragma unroll
            for (int ia = 0; ia < 4; ++ia) {
                const float* p = arow[ia] + k0 + 2 * half;
                a[ia].x = p[0];
                a[ia].y = p[1];
            }
#pragma unroll
            for (int jb = 0; jb < 4; ++jb) {
                const float* p = brow[jb] + k0 + 2 * half;
                bb[jb].x = p[0];
                bb[jb].y = p[1];
            }
#pragma unroll
            for (int ia = 0; ia < 4; ++ia)
#pragma unroll
                for (int jb = 0; jb < 4; ++jb)
                    acc[ia][jb] = wmma_f32_16x16x4(a[ia], bb[jb], acc[ia][jb]);
        }

#pragma unroll
        for (int ia = 0; ia < 4; ++ia)
#pragma unroll
            for (int jb = 0; jb < 4; ++jb)
#pragma unroll
                for (int v = 0; v < 8; ++v) {
                    const int srow = i0 + 16 * ia + v + 8 * half;
                    const int scol = j0 + 16 * jb + m;       // n = m
                    const int t = (scol - srow) & TMASK;
                    atomicAdd(&cl[t], acc[ia][jb][v]);       // ds_add_f32
                }
    }
    __syncthreads();

    float* cg = corr + (size_t)b * T_;
    for (int t = threadIdx.x; t < T_; t += blockDim.x)
        atomicAdd(&cg[t], cl[t] * (1.0f / (float)D_));
}

// ---------------------------------------------------------------------------
// K3: top-12 delays per batch via 12 rounds of block-wide argmax.
// ---------------------------------------------------------------------------
__global__ void k_topk(const float* __restrict__ corr, int* __restrict__ delays) {
    __shared__ float vals[T_];
    __shared__ float sval[256];
    __shared__ int   sidx[256];
    __shared__ int   best;
    const int b   = blockIdx.x;
    const int tid = threadIdx.x;

    for (int t = tid; t < T_; t += 256) vals[t] = corr[b * T_ + t];
    __syncthreads();

    for (int k = 0; k < KTOP; ++k) {
        float bv = NEG_INF;
        int   bi = 0;
        for (int t = tid; t < T_; t += 256) {
            const float v = vals[t];
            if (v > bv) { bv = v; bi = t; }
        }
        sval[tid] = bv;
        sidx[tid] = bi;
        __syncthreads();
        for (int s = 128; s > 0; s >>= 1) {
            if (tid < s && sval[tid + s] > sval[tid]) {
                sval[tid] = sval[tid + s];
                sidx[tid] = sidx[tid + s];
            }
            __syncthreads();
        }
        if (tid == 0) {
            best = sidx[0];
            delays[b * KTOP + k] = sidx[0];
        }
        __syncthreads();
        if (tid == 0) vals[best] = NEG_INF;
        __syncthreads();
    }
}

// ---------------------------------------------------------------------------
// K4: out[b,t,:] = x[b,t,:] + (1/12) * sum_k q[b, (t+delay_k) % T, :]
// float4-vectorized; q (64 MB) is fully L2-resident on MI455X (192 MB L2).
// ---------------------------------------------------------------------------
__global__ void k_gather_avg(const float* __restrict__ x,
                             const float* __restrict__ q,
                             const int* __restrict__ delays,
                             float* __restrict__ out) {
    const int DV = D_ / 4;
    const size_t i = (size_t)blockIdx.x * blockDim.x + threadIdx.x; // B*T*DV
    const int b      = (int)(i / ((size_t)T_ * DV));
    const size_t rem = i - (size_t)b * T_ * DV;
    const int t      = (int)(rem / DV);
    const int dv     = (int)(rem % DV);

    const float4* q4 = (const float4*)q + (size_t)b * T_ * DV;
    float4 acc = make_float4(0.f, 0.f, 0.f, 0.f);
#pragma unroll
    for (int k = 0; k < KTOP; ++k) {
        const int tt = (t + delays[b * KTOP + k]) & TMASK;
        const float4 v = q4[(size_t)tt * DV + dv];
        acc.x += v.x; acc.y += v.y; acc.z += v.z; acc.w += v.w;
    }
    const float4 xv = ((const float4*)x)[i];
    const float inv = 1.0f / (float)KTOP;
    float4 o;
    o.x = xv.x + acc.x * inv;
    o.y = xv.y + acc.y * inv;
    o.z = xv.z + acc.z * inv;
    o.w = xv.w + acc.w * inv;
    ((float4*)out)[i] = o;
}

// ---------------------------------------------------------------------------
extern "C" void kernel_launch(void* const* d_in, const int* in_sizes, int n_in,
                              void* d_out, int out_size, void* d_ws, size_t ws_size,
                              hipStream_t stream) {
    const float* x    = (const float*)d_in[0];  // [8, 4096, 512]
    const float* W    = (const float*)d_in[1];  // [512, 512]
    const float* bias = (const float*)d_in[2];  // [512]
    float* out = (float*)d_out;                 // [8, 4096, 512]

    // Workspace: q (B*T*D f32 = 64MB) | corr (B*T f32 = 128KB) | delays (B*12 i32)
    float* q      = (float*)d_ws;
    float* corr   = q + (size_t)B_ * T_ * D_;
    int*   delays = (int*)(corr + (size_t)B_ * T_);

    // K0: zero corr accumulator
    k_zero_corr<<<(B_ * T_ + 255) / 256, 256, 0, stream>>>(corr);

    // K1: projection GEMM (2048 row-tiles x 8 col-supertiles, 8 waves/block)
    k_proj_gemm<<<(2048 * 8) / 8, 256, 0, stream>>>(x, W, bias, q);

    // K2: Gram autocorrelation (one block per (i-supertile, batch))
    dim3 g2(T_ / 64, B_);
    k_gram_corr<<<g2, 256, 0, stream>>>(q, corr);

    // K3: top-12 per batch
    k_topk<<<B_, 256, 0, stream>>>(corr, delays);

    // K4: gather + average + residual
    k_gather_avg<<<(B_ * T_ * (D_ / 4)) / 256, 256, 0, stream>>>(x, q, delays, out);
}